// GraphNN_57827439673949
// MI455X (gfx1250) — compile-verified
//
#include <hip/hip_runtime.h>
#include <hip/hip_bf16.h>

// Problem constants (from reference)
static constexpr int   NV     = 10000;   // nodes
static constexpr int   NFEAT  = 128;
static constexpr int   NHID   = 3;
static constexpr int   GOUT   = 512;
static constexpr int   FCK    = NV * NHID;   // 30000
static constexpr float ALPHA_ = 0.2f;

typedef __attribute__((ext_vector_type(2))) float v2f;
typedef __attribute__((ext_vector_type(8))) float v8f;

// ---------------------------------------------------------------------------
// Kernel 1: Wh = emb @ W (10000x3), e_src = Wh@a[:3], e_dst = Wh@a[3:].
// Emits Wh padded to 16 floats/row: {Wh0, Wh1, Wh2, 1.0, 0 x 12} so it can be
// loaded directly as the WMMA B-matrix (column 3 accumulates the softmax
// denominator, columns 4..15 contribute exact zeros).
// ---------------------------------------------------------------------------
__global__ __launch_bounds__(256) void gat_prep(
    const float* __restrict__ emb, const float* __restrict__ W,
    const float* __restrict__ a, float* __restrict__ Wh16,
    float* __restrict__ es, float* __restrict__ ed) {
  int i = blockIdx.x * blockDim.x + threadIdx.x;
  if (i >= NV) return;
  const float* er = emb + (size_t)i * NFEAT;
  float h0 = 0.f, h1 = 0.f, h2 = 0.f;
  #pragma unroll 4
  for (int c = 0; c < NFEAT; ++c) {
    float e = er[c];                 // W[c][k] reads are wave-uniform -> SMEM
    h0 = fmaf(e, W[c * 3 + 0], h0);
    h1 = fmaf(e, W[c * 3 + 1], h1);
    h2 = fmaf(e, W[c * 3 + 2], h2);
  }
  float* o = Wh16 + (size_t)i * 16;
  o[0] = h0; o[1] = h1; o[2] = h2; o[3] = 1.0f;
  #pragma unroll
  for (int c = 4; c < 16; ++c) o[c] = 0.0f;
  es[i] = h0 * a[0] + h1 * a[1] + h2 * a[2];
  ed[i] = h0 * a[3] + h1 * a[4] + h2 * a[5];
}

// ---------------------------------------------------------------------------
// Kernel 2: fused masked-softmax attention + aggregation, one pass over adj.
// 16 rows per block; 8 waves split the j-dimension into contiguous chunks;
// each wave accumulates a 16x16 WMMA tile: D = P(16x4) x Wh16(4x16) + D with
// p_ij = adj ? exp(leaky_relu(e_src[i]+e_dst[j])) : 0.
// A layout per ISA: lane(0..15)=M, V0/V1 = K{0,1}; lanes 16..31 = K{2,3}.
// D columns 0..2 = unnormalized h, column 3 = sum of p (denominator).
// ---------------------------------------------------------------------------
__global__ __launch_bounds__(256) void gat_attn(
    const int* __restrict__ adj, const float* __restrict__ Wh16,
    const float* __restrict__ es, const float* __restrict__ ed,
    float* __restrict__ x) {
  __shared__ float red[8 * 64];

  const int tid  = threadIdx.x;
  const int lane = tid & 31;
  const int m    = lane & 15;    // M (row within 16-row tile)
  const int half = lane >> 4;    // 0: K=0,1   1: K=2,3
  const int kb   = half * 2;
  const int i0   = blockIdx.x * 16;
  const int row  = i0 + m;

  // wave id is uniform: make it explicit so the loop control is scalar and
  // EXEC is provably all-1s at every WMMA.
  const int wave = __builtin_amdgcn_readfirstlane(tid >> 5);
  const int IT   = NV / 4;                   // 2500 j-steps of 4
  const int lo   = (IT * wave) >> 3;         // contiguous chunk per wave
  const int hi   = (IT * (wave + 1)) >> 3;

  const float  esv  = es[row];
  const int*   arow = adj + (size_t)row * NV + kb;   // this lane's adj stream
  const float* edp  = ed + kb;
  const float* whp  = Wh16 + kb * 16 + m;            // B: lane m = column N=m

  v8f c = {};  // f32 16x16 accumulator (8 VGPRs / lane, wave32)

  #pragma unroll 2
  for (int it = lo; it < hi; ++it) {
    const int j0 = it * 4;
    // adj is 400MB read-once: one aligned 8B non-temporal load per lane
    // (index row*10000 + kb + j0 is always even -> 8B aligned).
    unsigned long long av =
        __builtin_nontemporal_load((const unsigned long long*)(arow + j0));
    int   a0 = (int)(unsigned)(av & 0xffffffffull);
    int   a1 = (int)(unsigned)(av >> 32);
    float e0 = edp[j0];
    float e1 = edp[j0 + 1];
    float b0 = whp[(size_t)j0 * 16];        // B row K=kb,   col m
    float b1 = whp[(size_t)j0 * 16 + 16];   // B row K=kb+1, col m
    float t0 = esv + e0;
    float t1 = esv + e1;
    t0 = fmaxf(t0, ALPHA_ * t0);            // leaky_relu (alpha < 1)
    t1 = fmaxf(t1, ALPHA_ * t1);
    float p0 = (a0 > 0) ? __expf(t0) : 0.0f;
    float p1 = (a1 > 0) ? __expf(t1) : 0.0f;
    v2f A = {p0, p1};
    v2f B = {b0, b1};
    c = __builtin_amdgcn_wmma_f32_16x16x4_f32(
        /*neg_a=*/false, A, /*neg_b=*/false, B,
        /*c_mod=*/(short)0, c, /*reuse_a=*/false, /*reuse_b=*/false);
  }

  // D layout: lane n (and n+16) hold column N=n; VGPR r -> M=r (+8 for upper).
  if (m < 4) {
    #pragma unroll
    for (int r = 0; r < 8; ++r) {
      int M = r + half * 8;
      red[wave * 64 + M * 4 + m] = c[r];
    }
  }
  __syncthreads();

  // Reduce the 8 wave partials: 64 values = 16 rows x {h0,h1,h2,sum}
  if (tid < 64) {
    float s = 0.0f;
    #pragma unroll
    for (int w = 0; w < 8; ++w) s += red[w * 64 + tid];
    red[tid] = s;   // each thread touches only its own slot
  }
  __syncthreads();

  // softmax normalize + ELU, write x (row-major 10000x3 flattened)
  if (tid < 64) {
    int M = tid >> 2, n = tid & 3;
    if (n < 3) {
      float S = red[(M << 2) | 3];
      S = (S == 0.0f) ? 1.0f : S;          // empty-row guard (prob ~ 0)
      float v = red[tid] / S;
      v = (v > 0.0f) ? v : (__expf(v) - 1.0f);   // ELU
      x[(size_t)(i0 + M) * NHID + n] = v;
    }
  }
}

// ---------------------------------------------------------------------------
// Kernel 3: out = x(1x30000) @ fc1_w.T(30000x512) + fc1_b. Pure bandwidth
// (61 MB of fc1_w); one block per output, coalesced along K.
// ---------------------------------------------------------------------------
__global__ __launch_bounds__(256) void gat_fc(
    const float* __restrict__ fw, const float* __restrict__ fb,
    const float* __restrict__ x, float* __restrict__ out) {
  __shared__ float red[256];
  const int o = blockIdx.x;
  const float* wr = fw + (size_t)o * FCK;
  float s = 0.0f;
  for (int t = threadIdx.x; t < FCK; t += 256)
    s = fmaf(__builtin_nontemporal_load(wr + t), x[t], s);
  red[threadIdx.x] = s;
  __syncthreads();
  for (int st = 128; st > 0; st >>= 1) {
    if (threadIdx.x < st) red[threadIdx.x] += red[threadIdx.x + st];
    __syncthreads();
  }
  if (threadIdx.x == 0) out[o] = red[0] + fb[o];
}

// ---------------------------------------------------------------------------
extern "C" void kernel_launch(void* const* d_in, const int* in_sizes, int n_in,
                              void* d_out, int out_size, void* d_ws, size_t ws_size,
                              hipStream_t stream) {
  const int*   adj = (const int*)d_in[0];
  const float* emb = (const float*)d_in[1];
  const float* W   = (const float*)d_in[2];
  const float* a   = (const float*)d_in[3];
  const float* fw  = (const float*)d_in[4];
  const float* fb  = (const float*)d_in[5];
  float* out = (float*)d_out;

  // Workspace: Wh16 (10000*16) | e_src (10000) | e_dst (10000) | x (30000)
  float* Wh16 = (float*)d_ws;
  float* es   = Wh16 + (size_t)NV * 16;
  float* ed   = es + NV;
  float* x    = ed + NV;

  gat_prep<<<(NV + 255) / 256, 256, 0, stream>>>(emb, W, a, Wh16, es, ed);
  gat_attn<<<NV / 16, 256, 0, stream>>>(adj, Wh16, es, ed, x);
  gat_fc<<<GOUT, 256, 0, stream>>>(fw, fb, x, out);
}